// RelationYOLOX_6296422056665
// MI455X (gfx1250) — compile-verified
//
#include <hip/hip_runtime.h>
#include <hip/hip_bf16.h>

// RelationYOLOX top-feature selection for MI455X (gfx1250, wave32).
//
// B=8, C=256, N=160*160=25600, K=1000.
// Pass 1: objectness GEMV via V_WMMA_F32_16X16X4_F32 (HBM-bound, ~210 MB).
// Pass 2: per-image top-K via 4-pass MSB radix-select + ballot-scan
//         compaction (L2-resident, latency-optimized: ~90 barriers total
//         instead of ~600).
// Pass 3: feature gather, coalesced writes.

typedef __attribute__((ext_vector_type(2))) float v2f;
typedef __attribute__((ext_vector_type(8))) float v8f;

#define BB    8
#define CC    256
#define NPOS  25600        // 160*160 == 25*1024
#define KSEL  1000
#define FGB   1.0e6f

// ---------------------------------------------------------------------------
// Kernel 1: objectness GEMV via WMMA f32 16x16x4 (one wave per 16 positions).
//   A[m][k] = roi[b][c0+k][n0+m], B[k][j] = w[c0+k] (broadcast columns),
//   D[m][j] = partial obj[n0+m] (identical across columns).
// A-layout (ISA 7.12.2): lanes 0-15 M=lane, vgpr0=K0/vgpr1=K1;
// lanes 16-31 M=lane-16, vgpr0=K2/vgpr1=K3. B mirrors the K split.
// D-layout: vgpr r -> M=r (lanes 0-15) / M=r+8 (lanes 16-31).
// ---------------------------------------------------------------------------
__global__ __launch_bounds__(256) void obj_wmma_kernel(
    const float* __restrict__ roi,   // [B][C][N]
    const int*   __restrict__ fg,    // [B][N]
    const float* __restrict__ w,     // [C]
    const float* __restrict__ bptr,  // [1]
    float*       __restrict__ obj,   // ws: [B][N]
    unsigned*    __restrict__ keys)  // ws: [B][N] monotone sort keys
{
    const int wave  = (blockIdx.x * blockDim.x + threadIdx.x) >> 5;
    const int lane  = threadIdx.x & 31;
    const int half  = lane >> 4;
    const int l     = lane & 15;
    const int tilesPerB = NPOS / 16;
    const int bimg = wave / tilesPerB;
    const int n0   = (wave % tilesPerB) * 16;

    const float* base = roi + ((size_t)bimg * CC) * NPOS + n0 + l;

    v8f acc = {};
#pragma unroll 4
    for (int c0 = 0; c0 < CC; c0 += 4) {
        const int cA = c0 + 2 * half;
        v2f a, bw;
        a.x  = base[(size_t)cA * NPOS];
        a.y  = base[(size_t)(cA + 1) * NPOS];
        bw.x = w[cA];
        bw.y = w[cA + 1];
        acc = __builtin_amdgcn_wmma_f32_16x16x4_f32(
            false, a, false, bw, (short)0, acc, false, false);
    }

    const float bias = bptr[0];
    if (l == 0) {   // lanes 0 and 16: rows 0-7 / 8-15 of the tile
        const int mbase = n0 + half * 8;
#pragma unroll
        for (int r = 0; r < 8; ++r) {
            const float  v   = acc[r] + bias;
            const size_t off = (size_t)bimg * NPOS + (mbase + r);
            obj[off] = v;
            const float sc = v + (fg[off] ? FGB : 0.0f);
            unsigned u = __float_as_uint(sc);
            u = (u & 0x80000000u) ? ~u : (u | 0x80000000u);
            keys[off] = u;
        }
    }
}

// ---------------------------------------------------------------------------
// Kernel 2: per-image top-K. One 1024-thread block (32 waves) per image.
//  Phase A: MSB-first 8-bit radix select -> exact threshold key `thr` and
//           eqNeeded (#threshold-equal elements to keep, lowest index first).
//  Phase B: ordered compaction over 25 chunks of 1024, in-wave prefixes via
//           wave32 ballot/popc, cross-wave via 32-entry serial scan.
// Emits indices in ascending order (reference sorts inds).
// ---------------------------------------------------------------------------
__global__ __launch_bounds__(1024) void select_kernel(
    const unsigned* __restrict__ keys,   // [B][N]
    const float*    __restrict__ obj,    // [B][N]
    const int*      __restrict__ fg,     // [B][N]
    int*            __restrict__ idx_ws, // ws: [B][K]
    float*          __restrict__ out_obj,
    float*          __restrict__ out_fg,
    float*          __restrict__ out_inds)
{
    const int b    = blockIdx.x;
    const int tid  = threadIdx.x;
    const int lane = tid & 31;
    const int wv   = tid >> 5;            // 0..31
    const unsigned* kb = keys + (size_t)b * NPOS;

    __shared__ unsigned s_hist[256];
    __shared__ unsigned s_sel[2];         // [selected digit, remaining k]
    __shared__ unsigned s_wgt[32];
    __shared__ unsigned s_weq[32];
    __shared__ unsigned s_totg, s_tote;

    // ---- Phase A: radix select (4 passes, MSB first) ----
    unsigned pref = 0;
    unsigned k    = KSEL;
    for (int shift = 24; shift >= 0; shift -= 8) {
        if (tid < 256) s_hist[tid] = 0;
        __syncthreads();
        const unsigned hmask = (shift == 24) ? 0u : (0xFFFFFFFFu << (shift + 8));
        for (int n = tid; n < NPOS; n += 1024) {
            const unsigned key = kb[n];
            if ((key & hmask) == pref)
                atomicAdd(&s_hist[(key >> shift) & 255u], 1u);
        }
        __syncthreads();
        if (tid == 0) {
            unsigned cum = 0;
            for (int d = 255; d >= 0; --d) {
                const unsigned h = s_hist[d];
                if (cum + h >= k) { s_sel[0] = (unsigned)d; s_sel[1] = k - cum; break; }
                cum += h;
            }
        }
        __syncthreads();
        pref |= (s_sel[0] << shift);
        k     = s_sel[1];
        __syncthreads();
    }
    const unsigned thr      = pref;        // exact K-th largest key
    const int      eqNeeded = (int)k;      // #equal-valued slots to fill

    // ---- Phase B: ordered compaction (NPOS == 25 * 1024 exactly) ----
    unsigned selBase = 0, eqBase = 0;
    const unsigned lmask = (1u << lane) - 1u;
    for (int base = 0; base < NPOS; base += 1024) {
        const int      n  = base + tid;
        const unsigned kk = kb[n];
        const bool gt = (kk > thr);
        const bool eq = (kk == thr);

        // in-wave exclusive prefixes via ballot
        const unsigned mg = (unsigned)__ballot(gt);
        const unsigned me = (unsigned)__ballot(eq);
        const unsigned gtInW = __popc(mg & lmask);
        const unsigned eqInW = __popc(me & lmask);
        if (lane == 0) { s_wgt[wv] = __popc(mg); s_weq[wv] = __popc(me); }
        __syncthreads();
        if (tid == 0) {   // exclusive scan of 32 wave totals
            unsigned ag = 0, ae = 0;
            for (int i = 0; i < 32; ++i) {
                const unsigned tg = s_wgt[i], te = s_weq[i];
                s_wgt[i] = ag; s_weq[i] = ae;
                ag += tg; ae += te;
            }
            s_totg = ag; s_tote = ae;
        }
        __syncthreads();
        const unsigned exclGt = s_wgt[wv] + gtInW;
        const unsigned exclEq = s_weq[wv] + eqInW;
        const unsigned totGt  = s_totg;
        const unsigned totEq  = s_tote;

        int remEq = eqNeeded - (int)eqBase; if (remEq < 0) remEq = 0;
        const int  qualBefore = ((int)exclEq < remEq) ? (int)exclEq : remEq;
        const bool sel = gt || (eq && ((int)(eqBase + exclEq) < eqNeeded));
        const unsigned pos = selBase + exclGt + (unsigned)qualBefore;

        if (sel) {
            const size_t o = (size_t)b * KSEL + pos;
            idx_ws[o]   = n;
            out_inds[o] = (float)n;
            out_obj[o]  = obj[(size_t)b * NPOS + n];
            out_fg[o]   = fg[(size_t)b * NPOS + n] ? 1.0f : 0.0f;
        }

        const int qualChunk = ((int)totEq < remEq) ? (int)totEq : remEq;
        selBase += totGt + (unsigned)qualChunk;
        eqBase  += totEq;
        __syncthreads();   // protect s_wgt/s_weq reuse next chunk
    }
}

// ---------------------------------------------------------------------------
// Kernel 3: gather selected rows. Scattered reads (layout-forced), coalesced
// 1 KB writes per block.
// ---------------------------------------------------------------------------
__global__ __launch_bounds__(256) void gather_kernel(
    const float* __restrict__ roi,       // [B][C][N]
    const int*   __restrict__ idx_ws,    // [B][K]
    float*       __restrict__ out_feats) // [B][K][C]
{
    const int i   = blockIdx.x;          // b*K + j
    const int b   = i / KSEL;
    const int idx = idx_ws[i];
    const int c   = threadIdx.x;
    out_feats[(size_t)i * CC + c] = roi[((size_t)b * CC + c) * NPOS + idx];
}

// ---------------------------------------------------------------------------
extern "C" void kernel_launch(void* const* d_in, const int* in_sizes, int n_in,
                              void* d_out, int out_size, void* d_ws, size_t ws_size,
                              hipStream_t stream)
{
    const float* roi  = (const float*)d_in[0];  // [8,256,160,160]
    const int*   fg   = (const int*)  d_in[1];  // [8,25600]
    const float* w    = (const float*)d_in[2];  // [1,256,1,1]
    const float* bias = (const float*)d_in[3];  // [1]
    // d_in[4] = num_to_refine (constant 1000)

    float*    obj_ws  = (float*)d_ws;                       // B*N floats
    unsigned* keys_ws = (unsigned*)(obj_ws + BB * NPOS);    // B*N uints
    int*      idx_ws  = (int*)(keys_ws + BB * NPOS);        // B*K ints

    float* out_feats = (float*)d_out;
    float* out_obj   = out_feats + (size_t)BB * KSEL * CC;
    float* out_fg    = out_obj   + (size_t)BB * KSEL;
    float* out_inds  = out_fg    + (size_t)BB * KSEL;

    const int totalWaves = BB * (NPOS / 16);        // 12800
    obj_wmma_kernel<<<totalWaves / 8, 256, 0, stream>>>(roi, fg, w, bias,
                                                        obj_ws, keys_ws);
    select_kernel<<<BB, 1024, 0, stream>>>(keys_ws, obj_ws, fg, idx_ws,
                                           out_obj, out_fg, out_inds);
    gather_kernel<<<BB * KSEL, 256, 0, stream>>>(roi, idx_ws, out_feats);
}